// Rand2dPatchShift_62474594287749
// MI455X (gfx1250) — compile-verified
//
#include <hip/hip_runtime.h>
#include <cstdint>

// Rand2dPatchShift for x:(B=16, T=16, H=14, W=14, c=768) f32.
// out[b,t,n,:] = x[b, (t - shift[n]) % 16, n, :]
//
// Pure streaming permutation along T: 0 FLOPs, 308 MB of traffic.
// Implemented via the CDNA5 async data mover: global -> LDS -> global with
// GLOBAL_LOAD_ASYNC_TO_LDS_B128 / GLOBAL_STORE_ASYNC_FROM_LDS_B128 (ASYNCcnt),
// 16 asyncs batched per wave for deep memory-level parallelism.

#define T_DIM 16
#define N_DIM 196
#define C4    192   // 768 floats / 4 per row (float4 granularity)

// Closed form of the Python shift schedule, including the sticky `inv` bug:
// inv becomes -1 permanently at i=11; the only (a==1,b==1) index before that
// is n==8 (w==h), which keeps s=0.
__device__ __forceinline__ int shift_of(int n) {
    int w = n % 7;
    int h = n / 7;
    int a = w % 3;
    int b = h % 3;
    if (a == 0) return (b == 0) ? -4 : ((b == 1) ? 1 : 2);
    if (a == 1) return (b == 0) ? -1 : ((b == 1) ? ((n == 8) ? 0 : -1) : 3);
    return (b == 0) ? -2 : ((b == 1) ? -3 : 4);
}

__global__ __launch_bounds__(C4) void
rand2d_patch_shift_async(const float* __restrict__ x, float* __restrict__ out) {
    // 16 rows x 192 float4 x 16B = 48 KB staging buffer (LDS/WGP = 320 KB).
    __shared__ float4 buf[T_DIM * C4];

    const int tid = threadIdx.x;     // 0..191 -> float4 lane within a row
    const int n   = blockIdx.x;      // patch index 0..195
    const int b   = blockIdx.y;      // batch 0..15
    const int s   = shift_of(n);     // scalar per block

    const unsigned long long xb = (unsigned long long)(uintptr_t)x;
    const unsigned long long ob = (unsigned long long)(uintptr_t)out;
    // Low 32 bits of the generic pointer == wave-relative LDS byte offset.
    const unsigned lds0 = (unsigned)(uintptr_t)(&buf[0]);

    // Phase 1: issue 16 async B128 loads (gathered along T) into LDS.
    // 16 x 512B = 8 KB outstanding per wave before the wait.
#pragma unroll
    for (int t = 0; t < T_DIM; ++t) {
        const int srcT = (t - s) & (T_DIM - 1);   // python-style mod 16
        const unsigned long long src =
            xb + ((((unsigned long long)((b * T_DIM + srcT) * N_DIM + n)) * C4 + tid) << 4);
        const unsigned lds = lds0 + (unsigned)((t * C4 + tid) << 4);
        asm volatile("global_load_async_to_lds_b128 %0, %1, off"
                     :: "v"(lds), "v"(src)
                     : "memory");
    }

    // Wait for all async loads of this wave (LDS now populated for our lanes).
    asm volatile("s_wait_asynccnt 0" ::: "memory");

    // Phase 2: issue 16 async B128 stores from LDS to the shifted output rows.
    // Each thread stores exactly the bytes it loaded -> no barrier needed.
#pragma unroll
    for (int t = 0; t < T_DIM; ++t) {
        const unsigned long long dst =
            ob + ((((unsigned long long)((b * T_DIM + t) * N_DIM + n)) * C4 + tid) << 4);
        const unsigned lds = lds0 + (unsigned)((t * C4 + tid) << 4);
        asm volatile("global_store_async_from_lds_b128 %0, %1, off"
                     :: "v"(dst), "v"(lds)
                     : "memory");
    }
    // S_ENDPGM performs an implicit wait-idle (all counters 0), so the async
    // stores are guaranteed complete before the wave retires.
}

extern "C" void kernel_launch(void* const* d_in, const int* in_sizes, int n_in,
                              void* d_out, int out_size, void* d_ws, size_t ws_size,
                              hipStream_t stream) {
    (void)in_sizes; (void)n_in; (void)out_size; (void)d_ws; (void)ws_size;
    const float* x = (const float*)d_in[0];
    float* out     = (float*)d_out;

    dim3 grid(N_DIM, 16);   // (n, b) = (196, 16) -> 3136 blocks
    dim3 block(C4);         // 192 threads = 6 wave32s
    rand2d_patch_shift_async<<<grid, block, 0, stream>>>(x, out);
}